// Conv2d_28802050687374
// MI455X (gfx1250) — compile-verified
//
#include <hip/hip_runtime.h>

// CDNA5 (gfx1250) 3x3 valid cross-correlation, f32, WMMA-based banded GEMM.
// Out(16x16 tile) = sum_{i=0..2} A_i(16x20) x Band_i(20x16), K split 5x4
// -> 15 v_wmma_f32_16x16x4_f32 per tile, one v8f accumulator.
typedef __attribute__((ext_vector_type(2))) float v2f;
typedef __attribute__((ext_vector_type(8))) float v8f;

#define TPW 8   // 16-row tiles per wave (amortizes Band setup)

__global__ __launch_bounds__(32)
void conv3x3_wmma_f32(const float* __restrict__ x,
                      const float* __restrict__ wgt,
                      float* __restrict__ out)
{
    constexpr int HH = 8192, WW = 8192;
    constexpr int OH = 8190, OW = 8190;

    const int lane = threadIdx.x & 31;
    const int half = lane >> 4;    // lane 16..31 -> 1
    const int lm   = lane & 15;    // M (for A/D) or N (for B/D) index
    const int c0   = blockIdx.x << 4;

    // 9 weights (uniform scalar loads)
    float w00 = wgt[0], w01 = wgt[1], w02 = wgt[2];
    float w10 = wgt[3], w11 = wgt[4], w12 = wgt[5];
    float w20 = wgt[6], w21 = wgt[7], w22 = wgt[8];
    float wv[3][3] = {{w00,w01,w02},{w10,w11,w12},{w20,w21,w22}};

    // Band operands. B f32 4x16 layout: vgpr v, lane-half h -> k = v + 2h, N = lane&15.
    // Chunk n covers global K = 4n + klocal; Band[k,c] = w[i][k-c] iff 0 <= k-c <= 2.
    v2f B[3][5];
#pragma unroll
    for (int i = 0; i < 3; ++i) {
#pragma unroll
        for (int n = 0; n < 5; ++n) {
            const int k0 = 4 * n + 2 * half;   // k for vgpr0
            const int d0 = k0 - lm;            // tap index for vgpr0
            const int d1 = d0 + 1;             // tap index for vgpr1
            float b0 = 0.0f, b1 = 0.0f;
            if (d0 == 0) b0 = wv[i][0]; else if (d0 == 1) b0 = wv[i][1]; else if (d0 == 2) b0 = wv[i][2];
            if (d1 == 0) b1 = wv[i][0]; else if (d1 == 1) b1 = wv[i][1]; else if (d1 == 2) b1 = wv[i][2];
            B[i][n].x = b0;
            B[i][n].y = b1;
        }
    }

    // Fast-path legality (column part, block-uniform):
    //   loads need c0+19 <= 8191, stores need c0+15 <= 8189  -> c0 <= 8172
    const bool col_fast = (c0 + 20 <= WW);

    int r0 = (blockIdx.y * TPW) << 4;
    for (int t = 0; t < TPW; ++t, r0 += 16) {
        v8f acc = {};  // C = 0

        // rows: loads need r0+17 <= 8191, stores need r0+15 <= 8189 -> r0 <= 8174
        const bool fast = col_fast & (r0 + 18 <= HH);

        if (fast) {
            // One per-lane base; all 15 A-loads are compile-time byte offsets
            // (i*WW + 4n floats), lowering to immediate-offset global_load_b64.
            const float* base = x + (size_t)(r0 + lm) * WW + (c0 + 2 * half);

            // Warm L2 for next tile's rows (speculative global_prefetch_b8).
            if (r0 + 32 <= HH)
                __builtin_prefetch((const void*)(base + (size_t)16 * WW), 0, 2);

#pragma unroll
            for (int i = 0; i < 3; ++i) {
#pragma unroll
                for (int n = 0; n < 5; ++n) {
                    const v2f a = *(const v2f*)(base + i * WW + 4 * n);
                    acc = __builtin_amdgcn_wmma_f32_16x16x4_f32(
                        false, a, false, B[i][n], (short)0, acc, false, false);
                }
            }

            // D layout: vgpr v -> row r0 + v + 8*half, col c0 + lm.
            float* obase = out + (size_t)(r0 + 8 * half) * OW + (c0 + lm);
#pragma unroll
            for (int v = 0; v < 8; ++v)
                obase[(size_t)v * OW] = acc[v];
        } else {
            // Edge tiles: clamp loads (garbage hits zero band rows or masked
            // stores), guard stores. EXEC stays all-1s through the WMMAs.
#pragma unroll
            for (int i = 0; i < 3; ++i) {
                int gr = r0 + i + lm;
                if (gr > HH - 1) gr = HH - 1;
                const float* rowp = x + (size_t)gr * WW;
#pragma unroll
                for (int n = 0; n < 5; ++n) {
                    int gc = c0 + 4 * n + 2 * half;
                    if (gc > WW - 2) gc = WW - 2;
                    const v2f a = *(const v2f*)(rowp + gc);
                    acc = __builtin_amdgcn_wmma_f32_16x16x4_f32(
                        false, a, false, B[i][n], (short)0, acc, false, false);
                }
            }
#pragma unroll
            for (int v = 0; v < 8; ++v) {
                const int orow = r0 + v + 8 * half;
                const int ocol = c0 + lm;
                if (orow < OH && ocol < OW)
                    out[(size_t)orow * OW + ocol] = acc[v];
            }
        }
    }
}

extern "C" void kernel_launch(void* const* d_in, const int* in_sizes, int n_in,
                              void* d_out, int out_size, void* d_ws, size_t ws_size,
                              hipStream_t stream) {
    const float* x   = (const float*)d_in[0];   // 8192*8192 f32
    const float* wgt = (const float*)d_in[1];   // 3*3 f32
    float* out       = (float*)d_out;           // 8190*8190 f32

    const int OW = 8190, OH = 8190;
    dim3 block(32, 1, 1);                             // one wave32 per block
    dim3 grid((OW + 15) / 16,                         // 512 tile columns
              (OH + 16 * TPW - 1) / (16 * TPW), 1);   // 64 row-groups of 8 tiles
    conv3x3_wmma_f32<<<grid, block, 0, stream>>>(x, wgt, out);
}